// Net_41403484733535
// MI455X (gfx1250) — compile-verified
//
#include <hip/hip_runtime.h>

typedef float v2f __attribute__((ext_vector_type(2)));
typedef float v8f __attribute__((ext_vector_type(8)));

#define DIMN 10
#define NPOS 100   // 10*10
#define NPAD 112   // 7 tiles * 16

// One wave32 per batch element.
//   Q(16x16 tile) = A(16x4) x B(4x16) accumulated over 3 K-chunks (K=12, 10 used)
//   A = W2 as (tap 9->16 pad) x (channel 10->12 pad)
//   B = T   as (channel chunk) x (16 positions), generated in-register
__global__ __launch_bounds__(32) void fused_net_kernel(
    const float* __restrict__ x,
    const float* __restrict__ w1,
    const float* __restrict__ b1,
    const float* __restrict__ w2,
    const float* __restrict__ b2,
    const float* __restrict__ fc1_w,
    const float* __restrict__ fc1_b,
    const float* __restrict__ fc2_w,
    const float* __restrict__ fc2_b,
    float* __restrict__ out)
{
    __shared__ float qlds[NPAD * 9];            // q[pos*9 + tap]
    __shared__ __align__(16) float ml[NPOS];    // x staging, then conv output map
    __shared__ float sbuf[20];                  // row sums [0..9], col sums [10..19]
    __shared__ float vl[DIMN];

    const int lane = threadIdx.x;   // 0..31
    const int bidx = blockIdx.x;    // batch element
    const float* xb = x + (size_t)bidx * NPOS;  // 400 B stride -> 16B aligned

    // ---- stage 0: stage x[b] through LDS (25 x float4) ----
    if (lane < 25) {
        const float4 v = ((const float4*)xb)[lane];
        ((float4*)ml)[lane] = v;
    }
    __syncthreads();

    // ---- stage 1: row / col sums ----
    if (lane < 20) {
        float s = 0.f;
        if (lane < DIMN) {
            #pragma unroll
            for (int j = 0; j < DIMN; ++j) s += ml[lane * DIMN + j];
        } else {
            const int j = lane - DIMN;
            #pragma unroll
            for (int i = 0; i < DIMN; ++i) s += ml[i * DIMN + j];
        }
        sbuf[lane] = s;
    }
    __syncthreads();

    // ---- per-lane WMMA operand constants ----
    const int  m  = lane & 15;        // M (tap) for A, N (pos-in-tile) for B
    const bool hi = lane >= 16;       // hi half handles K pair {2,3} of each chunk

    float w1r[3][2], b1r[3][2];
    v2f aW[3];
    #pragma unroll
    for (int kc3 = 0; kc3 < 3; ++kc3) {
        const int c0 = kc3 * 4 + (hi ? 2 : 0);
        const int c1 = c0 + 1;
        w1r[kc3][0] = (c0 < DIMN) ? w1[c0] : 0.f;
        w1r[kc3][1] = (c1 < DIMN) ? w1[c1] : 0.f;
        b1r[kc3][0] = (c0 < DIMN) ? b1[c0] : 0.f;
        b1r[kc3][1] = (c1 < DIMN) ? b1[c1] : 0.f;
        float a0 = 0.f, a1 = 0.f;
        if (m < 9) {                                  // 9 real taps, rows 9..15 zero
            if (c0 < DIMN) a0 = w2[c0 * 9 + m];       // w2[(1),c,a,b] flat: c*9 + tap
            if (c1 < DIMN) a1 = w2[c1 * 9 + m];
        }
        aW[kc3].x = a0; aW[kc3].y = a1;
    }

    // ---- hoist the s-values / masks for all 7 tiles (batch the LDS loads) ----
    float sv[7];
    float vmask[7];
    #pragma unroll
    for (int tile = 0; tile < 7; ++tile) {
        const int pos = tile * 16 + m;
        const int i   = pos / DIMN;               // up to 11; sbuf reads stay in-bounds
        const int j   = pos - i * DIMN;
        sv[tile]    = sbuf[i] + sbuf[DIMN + j];
        vmask[tile] = ((pos < NPOS) && (j <= i)) ? 1.f : 0.f;  // bounds + tril mask
    }

    // ---- stage 2: Q = W2 x T via 7 tiles of 16 positions ----
    #pragma unroll
    for (int tile = 0; tile < 7; ++tile) {
        const int   pos  = tile * 16 + m;
        const float s    = sv[tile];
        const float msk  = vmask[tile];

        v8f acc = {};
        #pragma unroll
        for (int kc3 = 0; kc3 < 3; ++kc3) {
            const float t0 = fmaxf(fmaf(w1r[kc3][0], s, b1r[kc3][0]), 0.f);
            const float t1 = fmaxf(fmaf(w1r[kc3][1], s, b1r[kc3][1]), 0.f);
            v2f bT;
            bT.x = t0 * msk;
            bT.y = t1 * msk;
            acc = __builtin_amdgcn_wmma_f32_16x16x4_f32(
                /*neg_a=*/false, aW[kc3], /*neg_b=*/false, bT,
                /*c_mod=*/(short)0, acc, /*reuse_a=*/false, /*reuse_b=*/false);
        }

        // C/D layout: VGPR r = tap r (lo lanes) / tap r+8 (hi lanes), N = m.
        if (!hi) {
            #pragma unroll
            for (int r = 0; r < 8; ++r) qlds[pos * 9 + r] = acc[r];
        } else {
            qlds[pos * 9 + 8] = acc[0];              // tap 8; taps 9..15 are zero
        }
    }
    __syncthreads();

    // ---- stage 3: 9-tap shifted gather + bias + relu -> conv map ----
    const float bias2 = b2[0];
    for (int p = lane; p < NPOS; p += 32) {
        const int i = p / DIMN;
        const int j = p - i * DIMN;
        float acc = bias2;
        #pragma unroll
        for (int a = 0; a < 3; ++a) {
            #pragma unroll
            for (int bb = 0; bb < 3; ++bb) {
                const int ii = i + a - 1;
                const int jj = j + bb - 1;
                if (ii >= 0 && ii < DIMN && jj >= 0 && jj < DIMN)
                    acc += qlds[(ii * DIMN + jj) * 9 + (a * 3 + bb)];
            }
        }
        ml[p] = fmaxf(acc, 0.f);
    }
    __syncthreads();

    // ---- stage 4: v[k] = row_sum(m)[k] + col_sum(m)[k] ----
    if (lane < DIMN) {
        float v = 0.f;
        #pragma unroll
        for (int j = 0; j < DIMN; ++j) v += ml[lane * DIMN + j];
        #pragma unroll
        for (int i = 0; i < DIMN; ++i) v += ml[i * DIMN + lane];
        vl[lane] = v;
    }
    __syncthreads();

    // ---- stage 5: fc1 (relu) + fc2 ----
    if (lane < 2) {
        float o = fc2_b[lane];
        #pragma unroll
        for (int u = 0; u < 4; ++u) {
            float hsum = fc1_b[u];
            #pragma unroll
            for (int k = 0; k < DIMN; ++k)
                hsum = fmaf(fc1_w[u * DIMN + k], vl[k], hsum);
            hsum = fmaxf(hsum, 0.f);
            o = fmaf(fc2_w[lane * 4 + u], hsum, o);
        }
        out[(size_t)bidx * 2 + lane] = o;
    }
}

extern "C" void kernel_launch(void* const* d_in, const int* in_sizes, int n_in,
                              void* d_out, int out_size, void* d_ws, size_t ws_size,
                              hipStream_t stream) {
    const float* x     = (const float*)d_in[0];
    const float* w1    = (const float*)d_in[1];
    const float* b1    = (const float*)d_in[2];
    const float* w2    = (const float*)d_in[3];
    const float* b2    = (const float*)d_in[4];
    const float* fc1_w = (const float*)d_in[5];
    const float* fc1_b = (const float*)d_in[6];
    const float* fc2_w = (const float*)d_in[7];
    const float* fc2_b = (const float*)d_in[8];
    float* out = (float*)d_out;

    const int batch = in_sizes[0] / NPOS;  // 65536
    fused_net_kernel<<<batch, 32, 0, stream>>>(x, w1, b1, w2, b2,
                                               fc1_w, fc1_b, fc2_w, fc2_b, out);
}